// DeepDCNN_53592601919804
// MI455X (gfx1250) — compile-verified
//
#include <hip/hip_runtime.h>
#include <hip/hip_bf16.h>
#include <math.h>

typedef __attribute__((ext_vector_type(2))) float v2f;
typedef __attribute__((ext_vector_type(8))) float v8f;

// ---------------------------------------------------------------------------
// Kernel 1: embedding gather + transpose -> x0 (B=64, C=64, S=1024), f32
// ---------------------------------------------------------------------------
__global__ __launch_bounds__(256) void embed_gather_k(
    const int* __restrict__ tokens, const float* __restrict__ emb,
    float* __restrict__ x0)
{
    int idx = blockIdx.x * 256 + threadIdx.x;   // 0 .. 64*64*1024-1
    int s = idx & 1023;
    int c = (idx >> 10) & 63;
    int b = idx >> 16;
    int tok = tokens[(b << 10) + s];
    x0[idx] = emb[tok * 64 + c];                // (b*64 + c)*1024 + s == idx
}

// ---------------------------------------------------------------------------
// Kernel 2: fused grouped-conv1d + bias + channel-pair fold via fp32 WMMA.
//
// Folded group g reads input channels [2g*IN_PG, 2g*IN_PG + CIN2).
// Per-channel tap count padded to a power of two (KTP) so the contraction
// index decomposes with shift/mask only:  c = k1 >> SHIFT, k = k1 & (KTP-1).
// Padded taps carry zero weight.
//
// Block = 128 threads (4 waves); each block owns one (b, g, 64-wide seq
// chunk). Input tile (CIN2 x TW, TW = 64+KSZ-1) is staged into LDS with
// CDNA5 async-to-LDS loads (ASYNCcnt + s_wait_asynccnt); halo zero-filled
// with ds_stores on disjoint addresses. Each wave then runs a fully
// unrolled K loop of V_WMMA_F32_16X16X4_F32 with all operands from LDS.
//
// A frag: lane m = lane&15, K = 2*(lane>>4) + v   (v = A vgpr 0/1)
// B frag: lane n = lane&15, K = 2*(lane>>4) + v
// C/D:    lane n = lane&15, M = r + 8*(lane>>4)   (r = 0..7)
// ---------------------------------------------------------------------------
template<int IN_PG, int KSZ, int NF, int G, int SIN, int SOUT, int CIN_TOTAL>
__global__ __launch_bounds__(128) void conv_fold_wmma_k(
    const float* __restrict__ x, const float* __restrict__ w,
    const float* __restrict__ bias, float* __restrict__ y, int chunks)
{
    constexpr int CIN2  = 2 * IN_PG;
    constexpr int KTP   = (KSZ <= 4) ? 4 : 8;       // taps padded per channel
    constexpr int SHIFT = (KTP == 4) ? 2 : 3;
    constexpr int KPAD  = CIN2 * KTP;               // padded contraction len
    constexpr int PAD   = KSZ - 1;
    constexpr int TW    = 64 + PAD;                 // staged tile width
    constexpr int COUT  = G * NF;

    __shared__ float sW[16 * KPAD];
    __shared__ __align__(16) float sX[CIN2 * TW + 8];  // +8: harmless pad-tap overread
    __shared__ float sB[16];

    const int tid   = threadIdx.x;
    const int chunk = blockIdx.x % chunks;
    const int g     = (blockIdx.x / chunks) % G;
    const int b     = blockIdx.x / (chunks * G);
    const int s0b   = chunk * 64;

    const float* xg = x + ((size_t)b * CIN_TOTAL + (size_t)(2 * g) * IN_PG) * SIN;

    // ---- stage input tile: async global->LDS for in-range, zero halo ----
    for (int idx = tid; idx < CIN2 * TW; idx += 128) {
        int c  = idx / TW;                 // compile-time TW -> mul/shift
        int sl = idx - c * TW;
        int sg = s0b - PAD + sl;
        if (sg >= 0 && sg < SIN) {
            const float* gp = xg + (size_t)c * SIN + sg;
            unsigned ldsoff = (unsigned)(uintptr_t)(&sX[idx]);  // low 32 = LDS addr
            asm volatile("global_load_async_to_lds_b32 %0, %1, off"
                         :: "v"(ldsoff), "v"(gp) : "memory");
        } else {
            sX[idx] = 0.f;
        }
    }

    // ---- stage fused (folded) weights, zero-padded taps / M rows ----
    #pragma unroll
    for (int m = 0; m < 16; ++m) {
        for (int k1 = tid; k1 < KPAD; k1 += 128) {
            int c = k1 >> SHIFT;
            int k = k1 & (KTP - 1);
            float val = 0.f;
            if (m < NF && k < KSZ) {
                int hi = (c >= IN_PG) ? 1 : 0;
                int gs = 2 * g + hi;
                int ic = c - hi * IN_PG;
                val = w[((size_t)(gs * NF + m) * IN_PG + ic) * KSZ + k];
            }
            sW[m * KPAD + k1] = val;
        }
    }
    if (tid < 16) {
        float bv = 0.f;
        if (tid < NF)
            bv = bias[(2 * g) * NF + tid] + bias[(2 * g + 1) * NF + tid];
        sB[tid] = bv;
    }

    asm volatile("s_wait_asynccnt 0" ::: "memory");
    __syncthreads();

    // ---- per-wave 16x16 output tile ----
    const int wave = tid >> 5;
    const int lane = tid & 31;
    const int s0   = s0b + wave * 16;
    if (s0 < SOUT) {                       // wave-uniform: EXEC all-1 inside
        const int n   = lane & 15;
        const int m16 = lane & 15;
        const int kh  = (lane >> 4) * 2;
        const int sw0 = m16 * KPAD + kh;           // A base offset
        const int sx0 = wave * 16 + n;             // in-tile seq position

        v8f acc = {0.f, 0.f, 0.f, 0.f, 0.f, 0.f, 0.f, 0.f};
        #pragma unroll
        for (int kk = 0; kk < KPAD; kk += 4) {
            const int k1 = kk + kh;                // even; pair stays in channel
            const int c  = k1 >> SHIFT;
            const int k  = k1 & (KTP - 1);
            v2f a, bb;
            a.x  = sW[sw0 + kk];
            a.y  = sW[sw0 + kk + 1];
            bb.x = sX[c * TW + sx0 + k];
            bb.y = sX[c * TW + sx0 + k + 1];
            acc = __builtin_amdgcn_wmma_f32_16x16x4_f32(
                      false, a, false, bb, (short)0, acc, false, false);
        }

        const int s = s0 + n;
        if (s < SOUT) {
            #pragma unroll
            for (int r = 0; r < 8; ++r) {
                int m = r + ((lane >> 4) << 3);
                if (m < NF)
                    y[((size_t)b * COUT + (size_t)(g * NF + m)) * SOUT + s] =
                        acc[r] + sB[m];
            }
        }
    }
}

// ---------------------------------------------------------------------------
// Kernel 3: exact order-preserving top-k + tanh per (b, channel) row.
// rank(i) = #{j: v[j] > v[i]} + #{j < i: v[j] == v[i]} (lax.top_k ties:
// lower index wins). Keep rank < k, compact in index order. Row in LDS,
// padded to x4 with -inf; candidate scan uses float4 LDS reads.
// ---------------------------------------------------------------------------
#define KMAX_MAXL 5
__global__ __launch_bounds__(256) void kmax_tanh_k(
    const float* __restrict__ y, float* __restrict__ xout, int n, int k)
{
    __shared__ __align__(16) float sv[1032];
    __shared__ unsigned char skeep[1032];
    __shared__ int           scs[256];

    const int    tid = threadIdx.x;
    const size_t row = blockIdx.x;
    const float* yr  = y + row * (size_t)n;
    const int    n4  = (n + 3) & ~3;

    for (int i = tid; i < n; i += 256) sv[i] = yr[i];
    for (int i = n + tid; i < n4; i += 256) sv[i] = -INFINITY;  // neutral pad
    __syncthreads();

    const int L  = (n + 255) / 256;            // <= 5 for all layers
    const int i0 = tid * L;
    const int i1 = (i0 + L < n) ? (i0 + L) : n;
    const int ne = (i1 > i0) ? (i1 - i0) : 0;

    float vi[KMAX_MAXL];
    int   rk[KMAX_MAXL];
    #pragma unroll
    for (int e = 0; e < KMAX_MAXL; ++e) {
        vi[e] = (e < ne) ? sv[i0 + e] : 0.f;
        rk[e] = 0;
    }
    for (int j4 = 0; j4 < n4; j4 += 4) {
        float4 vj = *(const float4*)&sv[j4];
        float  vja[4] = {vj.x, vj.y, vj.z, vj.w};
        #pragma unroll
        for (int t = 0; t < 4; ++t) {
            #pragma unroll
            for (int e = 0; e < KMAX_MAXL; ++e) {
                if (e < ne)
                    rk[e] += (vja[t] > vi[e]) ||
                             (vja[t] == vi[e] && (j4 + t) < i0 + e);
            }
        }
    }
    int cnt = 0;
    #pragma unroll
    for (int e = 0; e < KMAX_MAXL; ++e) {
        if (e < ne) {
            int kp = (rk[e] < k);
            skeep[i0 + e] = (unsigned char)kp;
            cnt += kp;
        }
    }
    scs[tid] = cnt;
    __syncthreads();
    if (tid == 0) {                            // serial exclusive scan (256)
        int run = 0;
        for (int t = 0; t < 256; ++t) { int c = scs[t]; scs[t] = run; run += c; }
    }
    __syncthreads();

    int    pos = scs[tid];
    float* xr  = xout + row * (size_t)k;
    for (int i = i0; i < i1; ++i) {
        if (skeep[i]) { xr[pos] = tanhf(sv[i]); ++pos; }
    }
}

// ---------------------------------------------------------------------------
// Kernel 4: FC (64 x 352) @ (6 x 352)^T + b  via fp32 WMMA.
// fcw staged zero-padded (16 x 352) in LDS so the K loop has no per-lane
// predication: unconditional ds_load pairs (B) + global b64 (A) + WMMA.
// 4 waves, each one 16(batch) x 16(classes, 6 valid) tile, K = 352.
// ---------------------------------------------------------------------------
__global__ __launch_bounds__(128) void fc_wmma_k(
    const float* __restrict__ x, const float* __restrict__ fcw,
    const float* __restrict__ fcb, float* __restrict__ out)
{
    __shared__ float sF[16 * 352];

    const int tid = threadIdx.x;

    // stage fcw rows 0..5, zero rows 6..15 (no divisions: loop per row)
    #pragma unroll
    for (int r = 0; r < 16; ++r) {
        for (int k1 = tid; k1 < 352; k1 += 128)
            sF[r * 352 + k1] = (r < 6) ? fcw[r * 352 + k1] : 0.f;
    }
    __syncthreads();

    const int wave = tid >> 5;
    const int lane = tid & 31;
    const int b0   = wave * 16;
    const int nn   = lane & 15;
    const int m16  = lane & 15;
    const int kh   = (lane >> 4) * 2;
    const float* xr = x + (size_t)(b0 + m16) * 352;
    const int    fb = nn * 352 + kh;

    v8f acc = {0.f, 0.f, 0.f, 0.f, 0.f, 0.f, 0.f, 0.f};
    #pragma unroll 8
    for (int kk = 0; kk < 352; kk += 4) {
        v2f a, bb;
        a.x  = xr[kk + kh];
        a.y  = xr[kk + kh + 1];
        bb.x = sF[fb + kk];
        bb.y = sF[fb + kk + 1];
        acc = __builtin_amdgcn_wmma_f32_16x16x4_f32(
                  false, a, false, bb, (short)0, acc, false, false);
    }
    if (nn < 6) {
        #pragma unroll
        for (int r = 0; r < 8; ++r) {
            int m = r + ((lane >> 4) << 3);
            out[(size_t)(b0 + m) * 6 + nn] = acc[r] + fcb[nn];
        }
    }
}

// ---------------------------------------------------------------------------
// Launch: embed -> 4x (conv_fold_wmma -> kmax_tanh) -> fc.
// Workspace ping-pong: X-buffer (max 15,728,640 f32 = x1) then Y-buffer
// (max 21,094,400 f32 = y0). Total ~147 MB.
// ---------------------------------------------------------------------------
extern "C" void kernel_launch(void* const* d_in, const int* in_sizes, int n_in,
                              void* d_out, int out_size, void* d_ws, size_t ws_size,
                              hipStream_t stream)
{
    const int*   tokens = (const int*)d_in[0];
    const float* emb    = (const float*)d_in[1];
    const float* w1 = (const float*)d_in[2],  *b1 = (const float*)d_in[3];
    const float* w2 = (const float*)d_in[4],  *b2 = (const float*)d_in[5];
    const float* w3 = (const float*)d_in[6],  *b3 = (const float*)d_in[7];
    const float* w4 = (const float*)d_in[8],  *b4 = (const float*)d_in[9];
    const float* fcw = (const float*)d_in[10];
    const float* fcb = (const float*)d_in[11];
    float*       out = (float*)d_out;

    float* bufX = (float*)d_ws;
    float* bufY = bufX + 15728640;   // max x-buffer (x1 = 64*320*768)

    // x0 = emb[tokens] transposed: (64, 64, 1024)
    embed_gather_k<<<16384, 256, 0, stream>>>(tokens, emb, bufX);

    // Layer 0: IN_PG=1,  K=7, NF=10, G=32, SIN=1024, SOUT=1030, CIN=64
    {
        const int chunks = (1030 + 63) / 64;                 // 17
        conv_fold_wmma_k<1, 7, 10, 32, 1024, 1030, 64>
            <<<64 * 32 * chunks, 128, 0, stream>>>(bufX, w1, b1, bufY, chunks);
        kmax_tanh_k<<<64 * 320, 256, 0, stream>>>(bufY, bufX, 1030, 768);
    }
    // Layer 1: IN_PG=10, K=5, NF=14, G=16, SIN=768, SOUT=772, CIN=320
    {
        const int chunks = (772 + 63) / 64;                  // 13
        conv_fold_wmma_k<10, 5, 14, 16, 768, 772, 320>
            <<<64 * 16 * chunks, 128, 0, stream>>>(bufX, w2, b2, bufY, chunks);
        kmax_tanh_k<<<64 * 224, 256, 0, stream>>>(bufY, bufX, 772, 512);
    }
    // Layer 2: IN_PG=14, K=5, NF=18, G=8, SIN=512, SOUT=516, CIN=224
    {
        const int chunks = (516 + 63) / 64;                  // 9
        conv_fold_wmma_k<14, 5, 18, 8, 512, 516, 224>
            <<<64 * 8 * chunks, 128, 0, stream>>>(bufX, w3, b3, bufY, chunks);
        kmax_tanh_k<<<64 * 144, 256, 0, stream>>>(bufY, bufX, 516, 256);
    }
    // Layer 3: IN_PG=18, K=3, NF=22, G=4, SIN=256, SOUT=258, CIN=144
    {
        const int chunks = (258 + 63) / 64;                  // 5
        conv_fold_wmma_k<18, 3, 22, 4, 256, 258, 144>
            <<<64 * 4 * chunks, 128, 0, stream>>>(bufX, w4, b4, bufY, chunks);
        kmax_tanh_k<<<64 * 88, 256, 0, stream>>>(bufY, bufX, 258, 4);
    }

    // bufX now holds (64, 88, 4) -> flattened (64, 352)
    fc_wmma_k<<<1, 128, 0, stream>>>(bufX, fcw, fcb, out);
}